// EMGOnsetCNN_84052509982971
// MI455X (gfx1250) — compile-verified
//
#include <hip/hip_runtime.h>
#include <math.h>

typedef __attribute__((ext_vector_type(2))) float v2f;
typedef __attribute__((ext_vector_type(8))) float v8f;
typedef __attribute__((ext_vector_type(4))) unsigned int u32x4;
typedef __attribute__((ext_vector_type(8))) int i32x8;
typedef __attribute__((ext_vector_type(4))) int i32x4;

#define T_LEN 8200
#define NF 129          // freq bins == frames
#define NPADB 272       // padded basis columns (258 -> 272)
#define CHUNK 32
#define NCHUNKS 8
#define PIX (NF*NF)     // 16641

#if __has_builtin(__builtin_amdgcn_tensor_load_to_lds) && __has_builtin(__builtin_amdgcn_s_wait_tensorcnt)
#define HAVE_TDM 1
#else
#define HAVE_TDM 0
#endif

#if HAVE_TDM
// 1-D TDM copy: nelem contiguous f32 from global -> LDS. Issued wave-level
// (EXEC ignored by TENSOR ops); caller guards so only one wave issues it,
// waits TENSORcnt==0, then a workgroup barrier publishes the LDS contents.
__device__ __forceinline__ void tdm_load_1d(const float* gsrc, void* lds_dst, int nelem) {
    unsigned long long ga = (unsigned long long)(uintptr_t)gsrc;
    unsigned int lds = (unsigned int)(uintptr_t)lds_dst;     // low 32 bits = LDS byte offset
    u32x4 g0;
    g0.x = 1u;                                               // count=1, is_restore=0
    g0.y = lds;                                              // lds_addr
    g0.z = (unsigned int)ga;                                 // global_addr[31:0]
    g0.w = (unsigned int)((ga >> 32) & 0x01FFFFFFu) | (2u << 30); // addr[56:32] | type=2
    unsigned int ne = (unsigned int)nelem;
    i32x8 g1;
    g1[0] = (int)(2u << 16);                 // workgroup_mask=0, data_size=2 (4 bytes)
    g1[1] = (int)(ne << 16);                 // tensor_dim0[15:0]  (bits 63:48)
    g1[2] = (int)((ne >> 16) | (1u << 16));  // tensor_dim0[31:16] | tensor_dim1=1
    g1[3] = (int)(ne << 16);                 // tile_dim0          (bits 127:112)
    g1[4] = 1;                               // tile_dim1=1, tile_dim2=0
    g1[5] = (int)ne;                         // tensor_dim0_stride[31:0]
    g1[6] = (int)((ne & 0xFFFFu) << 16);     // tensor_dim1_stride[15:0] (bits 223:208)
    g1[7] = (int)(ne >> 16);                 // tensor_dim1_stride[47:16]
    i32x4 z4 = {0, 0, 0, 0};
    i32x8 z8 = {0, 0, 0, 0, 0, 0, 0, 0};
    __builtin_amdgcn_tensor_load_to_lds(g0, g1, z4, z4, z8, 0);
    __builtin_amdgcn_s_wait_tensorcnt(0);
}
#endif

__device__ __forceinline__ int reflect_idx(int j) {
    if (j < 0) j = -j;
    if (j >= T_LEN) j = 2*(T_LEN-1) - j;
    return j;
}

// ---------------- basis: windowed DFT matrix (256 x 272) -----------------
__global__ void k_basis(float* __restrict__ basis) {
    int id = blockIdx.x*256 + threadIdx.x;
    if (id >= 256*NPADB) return;
    int k = id / NPADB, n = id - (id/NPADB)*NPADB;
    const float twopi = 6.283185307179586f;
    float win = 0.5f*(1.0f - cosf(twopi * (float)k / 256.0f));
    float v = 0.0f;
    if (n < 129)      v =  win * cosf(twopi * (float)n       * (float)k / 256.0f);
    else if (n < 258) v = -win * sinf(twopi * (float)(n-129) * (float)k / 256.0f);
    basis[id] = v;
}

// ---------------- weight transpose prepass: [oc][k] -> [k][oc] ------------
__global__ void k_wtrans(const float* __restrict__ w, float* __restrict__ wT,
                         int K, int OC) {
    int id = blockIdx.x*256 + threadIdx.x;
    if (id >= K*OC) return;
    int k = id / OC, oc = id - (id/OC)*OC;
    wT[id] = w[oc*K + k];
}

// ---------------- STFT GEMM: (32*129 x 256) x (256 x 272) via WMMA f32 ----
__global__ void k_stft_gemm(const float* __restrict__ x, const float* __restrict__ basis,
                            float* __restrict__ Ctmp, int b0) {
    int mt = blockIdx.x;            // 0..257  (4128/16)
    int nt = blockIdx.y;            // 0..16   (272/16)
    int lane = threadIdx.x;         // wave32
    int lh = lane >> 4, ll = lane & 15;
    int m = mt*16 + ll;             // frame-row: b_local*129 + t
    int bl = m / NF;
    int t  = m - bl*NF;
    const float* xb = x + (size_t)(b0 + bl) * T_LEN;
    int ncol = nt*16 + ll;
    v8f acc = {};
    for (int kb = 0; kb < 256; kb += 4) {
        int k0 = kb + 2*lh;
        v2f a, bv;
        int i0 = t*64 + k0 - 128;               // reflect-padded sample index
        a.x = xb[reflect_idx(i0)];
        a.y = xb[reflect_idx(i0 + 1)];
        bv.x = basis[(size_t)(kb + 0 + 2*lh)*NPADB + ncol];
        bv.y = basis[(size_t)(kb + 1 + 2*lh)*NPADB + ncol];
        acc = __builtin_amdgcn_wmma_f32_16x16x4_f32(false, a, false, bv,
                                                    (short)0, acc, false, false);
    }
    float* Cb = Ctmp + (size_t)(mt*16)*NPADB + nt*16 + ll;
    #pragma unroll
    for (int v = 0; v < 8; ++v)
        Cb[(size_t)(v + 8*lh)*NPADB] = acc[v];
}

// ---------------- combine re/im -> power spectrum s[b][f][t] --------------
__global__ void k_combine(const float* __restrict__ C, float* __restrict__ s) {
    int id = blockIdx.x*256 + threadIdx.x;      // (b, f, t) with t fastest
    if (id >= CHUNK*PIX) return;
    int b = id / PIX;
    int r = id - b*PIX;
    int f = r / NF, t = r - (r/NF)*NF;
    size_t mrow = (size_t)(b*NF + t) * NPADB;
    float re = C[mrow + f];
    float im = C[mrow + 129 + f];
    s[id] = (re*re + im*im) * (1.0f/96.0f);     // / sum(win^2)
}

// ---------------- conv1 1->16 + BN + ReLU (direct) ------------------------
__global__ void k_conv1(const float* __restrict__ s, const float* __restrict__ w,
                        const float* __restrict__ cb, const float* __restrict__ g,
                        const float* __restrict__ be, const float* __restrict__ mn,
                        const float* __restrict__ vr, float* __restrict__ h1) {
    __shared__ float wl[144];
    int tid = threadIdx.x;
    if (tid < 144) wl[tid] = w[tid];
    __syncthreads();
    int id = blockIdx.x*256 + tid;
    if (id >= CHUNK*PIX) return;
    int b = id / PIX;
    int r = id - b*PIX;
    int y = r / NF, xx = r - (r/NF)*NF;
    float in[9];
    #pragma unroll
    for (int tap = 0; tap < 9; ++tap) {
        int dy = tap/3 - 1, dx = tap%3 - 1;
        int yy = y + dy, xq = xx + dx;
        in[tap] = (yy >= 0 && yy < NF && xq >= 0 && xq < NF)
                  ? s[((size_t)b*NF + yy)*NF + xq] : 0.0f;
    }
    #pragma unroll
    for (int oc = 0; oc < 16; ++oc) {
        float acc = 0.0f;
        #pragma unroll
        for (int tap = 0; tap < 9; ++tap) acc += in[tap]*wl[oc*9 + tap];
        float sc = g[oc] * rsqrtf(vr[oc] + 1e-5f);
        float sh = (cb[oc] - mn[oc])*sc + be[oc];
        float z = acc*sc + sh;
        h1[(((size_t)b*16 + oc)*NF + y)*NF + xx] = fmaxf(z, 0.0f);
    }
}

// ---------------- conv2 16->32 + BN + ReLU via WMMA implicit-GEMM ---------
// weights arrive pre-transposed as wT[k*32 + oc], staged to LDS via TDM.
__global__ void k_conv2(const float* __restrict__ h1, const float* __restrict__ wT,
                        const float* __restrict__ cb, const float* __restrict__ g,
                        const float* __restrict__ be, const float* __restrict__ mn,
                        const float* __restrict__ vr, float* __restrict__ h2) {
    __shared__ float wl[144*32];                 // [k][oc]
    int tid = threadIdx.x;                       // 256 = 8 waves
#if HAVE_TDM
    if (tid == 0) tdm_load_1d(wT, (void*)wl, 144*32);
#else
    for (int i = tid; i < 144*32; i += 256) wl[i] = wT[i];
#endif
    __syncthreads();
    int b = blockIdx.x, yt = blockIdx.y, xt = blockIdx.z;
    int wv = tid >> 5, lane = tid & 31;
    int y = yt*8 + wv;
    if (y >= NF) return;                         // wave-uniform exit
    int lh = lane >> 4, ll = lane & 15;
    int xA = xt*16 + ll;
    const float* h1b = h1 + (size_t)b*16*PIX;
    __builtin_prefetch(h1b + (size_t)y*NF, 0, 1);   // global_prefetch_b8
    v8f acc0 = {}, acc1 = {};
    for (int kb = 0; kb < 144; kb += 4) {
        v2f a, bv0, bv1;
        #pragma unroll
        for (int v = 0; v < 2; ++v) {
            int k = kb + 2*lh + v;
            int c = k / 9, tap = k - c*9;
            int dy = tap/3 - 1, dx = tap%3 - 1;
            int yy = y + dy, xq = xA + dx;
            float val = 0.0f;
            if (yy >= 0 && yy < NF && xq >= 0 && xq < NF)
                val = h1b[((size_t)c*NF + yy)*NF + xq];
            a[v] = val;
            bv0[v] = wl[k*32 + ll];
            bv1[v] = wl[k*32 + 16 + ll];
        }
        acc0 = __builtin_amdgcn_wmma_f32_16x16x4_f32(false, a, false, bv0,
                                                     (short)0, acc0, false, false);
        acc1 = __builtin_amdgcn_wmma_f32_16x16x4_f32(false, a, false, bv1,
                                                     (short)0, acc1, false, false);
    }
    float* h2b = h2 + (size_t)b*32*PIX;
    v8f accs[2] = {acc0, acc1};
    #pragma unroll
    for (int half = 0; half < 2; ++half) {
        int oc = half*16 + ll;
        float sc = g[oc] * rsqrtf(vr[oc] + 1e-5f);
        float sh = (cb[oc] - mn[oc])*sc + be[oc];
        #pragma unroll
        for (int v = 0; v < 8; ++v) {
            int xp = xt*16 + v + 8*lh;
            if (xp < NF) {
                float z = accs[half][v]*sc + sh;
                h2b[((size_t)oc*NF + y)*NF + xp] = fmaxf(z, 0.0f);
            }
        }
    }
}

// ---------------- conv3 32->16 + BN + ReLU via WMMA implicit-GEMM ---------
__global__ void k_conv3(const float* __restrict__ h2, const float* __restrict__ wT,
                        const float* __restrict__ cb, const float* __restrict__ g,
                        const float* __restrict__ be, const float* __restrict__ mn,
                        const float* __restrict__ vr, float* __restrict__ h3) {
    __shared__ float wl[288*16];                 // [k][oc]
    int tid = threadIdx.x;
#if HAVE_TDM
    if (tid == 0) tdm_load_1d(wT, (void*)wl, 288*16);
#else
    for (int i = tid; i < 288*16; i += 256) wl[i] = wT[i];
#endif
    __syncthreads();
    int b = blockIdx.x, yt = blockIdx.y, xt = blockIdx.z;
    int wv = tid >> 5, lane = tid & 31;
    int y = yt*8 + wv;
    if (y >= NF) return;
    int lh = lane >> 4, ll = lane & 15;
    int xA = xt*16 + ll;
    const float* h2b = h2 + (size_t)b*32*PIX;
    v8f acc = {};
    for (int kb = 0; kb < 288; kb += 4) {
        v2f a, bv;
        #pragma unroll
        for (int v = 0; v < 2; ++v) {
            int k = kb + 2*lh + v;
            int c = k / 9, tap = k - c*9;
            int dy = tap/3 - 1, dx = tap%3 - 1;
            int yy = y + dy, xq = xA + dx;
            float val = 0.0f;
            if (yy >= 0 && yy < NF && xq >= 0 && xq < NF)
                val = h2b[((size_t)c*NF + yy)*NF + xq];
            a[v] = val;
            bv[v] = wl[k*16 + ll];
        }
        acc = __builtin_amdgcn_wmma_f32_16x16x4_f32(false, a, false, bv,
                                                    (short)0, acc, false, false);
    }
    float* h3b = h3 + (size_t)b*16*PIX;
    int oc = ll;
    float sc = g[oc] * rsqrtf(vr[oc] + 1e-5f);
    float sh = (cb[oc] - mn[oc])*sc + be[oc];
    #pragma unroll
    for (int v = 0; v < 8; ++v) {
        int xp = xt*16 + v + 8*lh;
        if (xp < NF) {
            float z = acc[v]*sc + sh;
            h3b[((size_t)oc*NF + y)*NF + xp] = fmaxf(z, 0.0f);
        }
    }
}

// ---------------- dec 16->1 (direct) --------------------------------------
__global__ void k_dec(const float* __restrict__ h3, const float* __restrict__ w,
                      const float* __restrict__ cb, float* __restrict__ dec) {
    __shared__ float wl[144];
    int tid = threadIdx.x;
    if (tid < 144) wl[tid] = w[tid];
    __syncthreads();
    int id = blockIdx.x*256 + tid;
    if (id >= CHUNK*PIX) return;
    int b = id / PIX;
    int r = id - b*PIX;
    int y = r / NF, xx = r - (r/NF)*NF;
    const float* h3b = h3 + (size_t)b*16*PIX;
    float acc = cb[0];
    #pragma unroll
    for (int c = 0; c < 16; ++c) {
        #pragma unroll
        for (int tap = 0; tap < 9; ++tap) {
            int dy = tap/3 - 1, dx = tap%3 - 1;
            int yy = y + dy, xq = xx + dx;
            if (yy >= 0 && yy < NF && xq >= 0 && xq < NF)
                acc += h3b[((size_t)c*NF + yy)*NF + xq] * wl[c*9 + tap];
        }
    }
    dec[id] = acc;
}

// ---------------- conv1d over freq, 129 "channels" = frames ---------------
__global__ void k_fin(const float* __restrict__ dec, const float* __restrict__ fw,
                      const float* __restrict__ fb, float* __restrict__ vout) {
    __shared__ float wl[129*3];
    int tid = threadIdx.x;
    for (int i = tid; i < 129*3; i += 128) wl[i] = fw[i];
    __syncthreads();
    int id = blockIdx.x*128 + tid;               // (b, f)
    if (id >= CHUNK*NF) return;
    int b = id / NF, f = id - (id/NF)*NF;
    const float* db = dec + (size_t)b*PIX;
    float acc = fb[0];
    for (int t = 0; t < NF; ++t) {
        float w0 = wl[t*3+0], w1 = wl[t*3+1], w2 = wl[t*3+2];
        if (f >= 1)    acc += db[(size_t)(f-1)*NF + t]*w0;
        acc += db[(size_t)f*NF + t]*w1;
        if (f+1 < NF)  acc += db[(size_t)(f+1)*NF + t]*w2;
    }
    vout[id] = acc;
}

// ---------------- linear interp 129 -> 8200 + sigmoid ---------------------
__global__ void k_interp(const float* __restrict__ vin, float* __restrict__ out, int b0) {
    int id = blockIdx.x*256 + threadIdx.x;       // (b, i)
    if (id >= CHUNK*T_LEN) return;
    int b = id / T_LEN, i = id - b*T_LEN;
    float pos = ((float)i + 0.5f)*(129.0f/8200.0f) - 0.5f;
    pos = fminf(fmaxf(pos, 0.0f), 128.0f);
    int lo = (int)floorf(pos);
    int hi = lo + 1 < 129 ? lo + 1 : 128;
    float wgt = pos - (float)lo;
    float v = vin[b*NF + lo]*(1.0f - wgt) + vin[b*NF + hi]*wgt;
    out[(size_t)(b0 + b)*T_LEN + i] = 1.0f/(1.0f + expf(-v));
}

extern "C" void kernel_launch(void* const* d_in, const int* in_sizes, int n_in,
                              void* d_out, int out_size, void* d_ws, size_t ws_size,
                              hipStream_t stream) {
    const float* x   = (const float*)d_in[0];
    const float* w1  = (const float*)d_in[1];  const float* b1 = (const float*)d_in[2];
    const float* g1  = (const float*)d_in[3];  const float* e1 = (const float*)d_in[4];
    const float* m1  = (const float*)d_in[5];  const float* v1 = (const float*)d_in[6];
    const float* w2  = (const float*)d_in[7];  const float* b2 = (const float*)d_in[8];
    const float* g2  = (const float*)d_in[9];  const float* e2 = (const float*)d_in[10];
    const float* m2  = (const float*)d_in[11]; const float* v2 = (const float*)d_in[12];
    const float* w3  = (const float*)d_in[13]; const float* b3 = (const float*)d_in[14];
    const float* g3  = (const float*)d_in[15]; const float* e3 = (const float*)d_in[16];
    const float* m3  = (const float*)d_in[17]; const float* v3 = (const float*)d_in[18];
    const float* wd  = (const float*)d_in[19]; const float* bd = (const float*)d_in[20];
    const float* wf  = (const float*)d_in[21]; const float* bf = (const float*)d_in[22];
    float* out = (float*)d_out;

    char* ws = (char*)d_ws;
    float* basis = (float*)(ws + 0x0);        // 256*272*4      = 278,528 B
    float* sbuf  = (float*)(ws + 0x50000);    // 32*129*129*4   = 2,130,048 B (also dec)
    float* h1buf = (float*)(ws + 0x260000);   // 32*16*129*129*4= 34,080,768 B (also h3)
    float* h2buf = (float*)(ws + 0x2300000);  // 32*32*129*129*4= 68,161,536 B (also Ctmp)
    float* vbuf  = (float*)(ws + 0x6500000);  // 32*129*4
    float* wT2   = (float*)(ws + 0x6510000);  // 144*32*4 = 18,432 B
    float* wT3   = (float*)(ws + 0x6520000);  // 288*16*4 = 18,432 B
    float* Ctmp  = h2buf;                     // 4128*272*4 = 4,491,264 B (before conv2)
    float* decb  = sbuf;                      // reused after conv1 consumed s

    k_basis<<<(256*NPADB + 255)/256, 256, 0, stream>>>(basis);
    k_wtrans<<<(144*32 + 255)/256, 256, 0, stream>>>(w2, wT2, 144, 32);
    k_wtrans<<<(288*16 + 255)/256, 256, 0, stream>>>(w3, wT3, 288, 16);

    for (int ch = 0; ch < NCHUNKS; ++ch) {
        int b0 = ch * CHUNK;
        k_stft_gemm<<<dim3(258, 17), 32, 0, stream>>>(x, basis, Ctmp, b0);
        k_combine<<<(CHUNK*PIX + 255)/256, 256, 0, stream>>>(Ctmp, sbuf);
        k_conv1<<<(CHUNK*PIX + 255)/256, 256, 0, stream>>>(sbuf, w1, b1, g1, e1, m1, v1, h1buf);
        k_conv2<<<dim3(CHUNK, 17, 9), 256, 0, stream>>>(h1buf, wT2, b2, g2, e2, m2, v2, h2buf);
        k_conv3<<<dim3(CHUNK, 17, 9), 256, 0, stream>>>(h2buf, wT3, b3, g3, e3, m3, v3, h1buf);
        k_dec<<<(CHUNK*PIX + 255)/256, 256, 0, stream>>>(h1buf, wd, bd, decb);
        k_fin<<<(CHUNK*NF + 127)/128, 128, 0, stream>>>(decb, wf, bf, vbuf);
        k_interp<<<(CHUNK*T_LEN + 255)/256, 256, 0, stream>>>(vbuf, out, b0);
    }
}